// RandomWalkPixelLabeling_3393024163841
// MI455X (gfx1250) — compile-verified
//
#include <hip/hip_runtime.h>
#include <hip/hip_bf16.h>
#include <stdint.h>

#define HH 96
#define WW 96
#define NPIX 9216          // 96*96
#define NT 1024            // threads in kernel A
#define PT 9               // pixels per thread in kernel A (9*1024 = 9216)
#define COEF 5.0f
#define BIGV 1000000000.0f

typedef float v4f __attribute__((ext_vector_type(4)));

// ---------------------------------------------------------------------------
// Kernel A (single workgroup, 32 wave32s on one WGP):
//  - async-load the 96x96 image into LDS (CDNA5 GLOBAL_LOAD_ASYNC_TO_LDS_B32)
//  - build the per-pixel "high" mask in LDS (byte array)
//  - compute flat_ per pixel (5x5 masked max, 3x3 masked min)
//  - overwrite LDS image with flat_, reduce diff max/min over all 4-neighbor
//    edges, emit scalars. Ordering: s_wait_xcnt + s_wait_asynccnt + barrier.
// ---------------------------------------------------------------------------
__global__ void __launch_bounds__(NT)
rwpl_flat_kernel(const float* __restrict__ img,
                 const float* __restrict__ thre,
                 const float* __restrict__ addv,
                 float* __restrict__ ws_flat,   // [NPIX]
                 float* __restrict__ ws_scal)   // [2] = {diffmax, diffmin}
{
    __shared__ float s_img[NPIX];               // 36 KB
    __shared__ unsigned char s_high[NPIX];      //  9 KB
    __shared__ float s_rmax[32];
    __shared__ float s_rmin[32];

    const unsigned t = threadIdx.x;

    // ---- async DMA of the image into LDS (tracked by ASYNCcnt) ----
    const unsigned lds_img = (unsigned)(size_t)(&s_img[0]); // low 32 bits of flat addr = LDS offset
#pragma unroll
    for (int k = 0; k < PT; ++k) {
        const unsigned off = (t + (unsigned)k * NT) * 4u;
        asm volatile("global_load_async_to_lds_b32 %0, %1, %2"
                     :
                     : "v"(lds_img + off), "v"(off), "s"(img)
                     : "memory");
    }
    // make address-VGPR recycling safe even under XNACK replay
    asm volatile("s_wait_xcnt 0x0" ::: "memory");

    // ---- meanwhile: normal loads of thresholds/additions into registers ----
    float myth[PT];
    float myadd[PT];
#pragma unroll
    for (int k = 0; k < PT; ++k) {
        const int p = (int)t + k * NT;
        myth[k]  = thre[p];
        myadd[k] = addv[p];
    }

    // wait for our wave's async LDS writes, then workgroup barrier
    asm volatile("s_wait_asynccnt 0x0" ::: "memory");
    __syncthreads();

#pragma unroll
    for (int k = 0; k < PT; ++k) {
        const int p = (int)t + k * NT;
        s_high[p] = (s_img[p] >= myth[k]) ? 1u : 0u;
    }
    __syncthreads();

    // ---- per-pixel flat_ ----
    float fl[PT];
#pragma unroll
    for (int k = 0; k < PT; ++k) {
        const int p = (int)t + k * NT;
        const int i = p / WW, j = p % WW;
        const float f    = s_img[p];
        const float high = s_high[p] ? 1.0f : 0.0f;

        float maxl = -BIGV;   // 5x5 local max over low-masked pixels
        float minl =  BIGV;   // 3x3 local min over high-masked pixels
        for (int di = -2; di <= 2; ++di) {
            const int ii = i + di;
            if (ii < 0 || ii >= HH) continue;
            for (int dj = -2; dj <= 2; ++dj) {
                const int jj = j + dj;
                if (jj < 0 || jj >= WW) continue;
                const int q = ii * WW + jj;
                const float fq = s_img[q];
                const bool hq = s_high[q] != 0;
                if (!hq) maxl = fmaxf(maxl, fq);
                if (hq && di >= -1 && di <= 1 && dj >= -1 && dj <= 1)
                    minl = fminf(minl, fq);
            }
        }
        if (maxl <= -BIGV * 0.5f) maxl = 0.0f;
        if (minl >=  BIGV * 0.5f) minl = 0.0f;

        const float img_high = (high * f - minl) / (1.0f - minl + 1e-11f);
        const float img_low  = ((1.0f - high) * f) / (maxl + 1e-11f);
        fl[k] = img_high + img_low + myadd[k] * high;
    }
    __syncthreads();

    // overwrite LDS image with flat_, also spill to global for the scatter kernel
#pragma unroll
    for (int k = 0; k < PT; ++k) {
        const int p = (int)t + k * NT;
        s_img[p]   = fl[k];
        ws_flat[p] = fl[k];
    }
    __syncthreads();

    // ---- diff max/min over all 4-neighbor edges (right+down covers each) ----
    float dmax = 0.0f, dmin = BIGV;
#pragma unroll
    for (int k = 0; k < PT; ++k) {
        const int p = (int)t + k * NT;
        const int i = p / WW, j = p % WW;
        const float fp = s_img[p];
        if (j + 1 < WW) {
            const float d = fabsf(fp - s_img[p + 1]);
            dmax = fmaxf(dmax, d); dmin = fminf(dmin, d);
        }
        if (i + 1 < HH) {
            const float d = fabsf(fp - s_img[p + WW]);
            dmax = fmaxf(dmax, d); dmin = fminf(dmin, d);
        }
    }
    // wave32 shuffle reduction
#pragma unroll
    for (int o = 16; o >= 1; o >>= 1) {
        dmax = fmaxf(dmax, __shfl_xor(dmax, o, 32));
        dmin = fminf(dmin, __shfl_xor(dmin, o, 32));
    }
    const int wave = (int)t >> 5, lane = (int)t & 31;
    if (lane == 0) { s_rmax[wave] = dmax; s_rmin[wave] = dmin; }
    __syncthreads();
    if (wave == 0) {
        dmax = s_rmax[lane];
        dmin = s_rmin[lane];
#pragma unroll
        for (int o = 16; o >= 1; o >>= 1) {
            dmax = fmaxf(dmax, __shfl_xor(dmax, o, 32));
            dmin = fminf(dmin, __shfl_xor(dmin, o, 32));
        }
        if (lane == 0) { ws_scal[0] = dmax; ws_scal[1] = dmin; }
    }
}

// ---------------------------------------------------------------------------
// Kernel B: stream 340 MB of zeros with nontemporal 128-bit stores.
// Output (340 MB) exceeds the 192 MB L2, so NT (bypass/rinse) is the right
// temporal hint; this is the roofline-dominant phase (~15 us @ 23.3 TB/s).
// 4x unrolled grid-stride loop; every global_store_b128 th:NT stays
// per-instruction coalesced (512 B per wave32 store).
// ---------------------------------------------------------------------------
__global__ void __launch_bounds__(256)
rwpl_fill_kernel(v4f* __restrict__ out, long long n4)
{
    const long long stride  = (long long)gridDim.x * blockDim.x;
    long long idx           = (long long)blockIdx.x * blockDim.x + threadIdx.x;
    const v4f z = {0.0f, 0.0f, 0.0f, 0.0f};

    const long long stride4 = stride * 4;
    long long limit = n4 - 3 * stride;
    for (; idx < limit; idx += stride4) {
        __builtin_nontemporal_store(z, &out[idx]);
        __builtin_nontemporal_store(z, &out[idx + stride]);
        __builtin_nontemporal_store(z, &out[idx + 2 * stride]);
        __builtin_nontemporal_store(z, &out[idx + 3 * stride]);
    }
    for (; idx < n4; idx += stride)
        __builtin_nontemporal_store(z, &out[idx]);
}

// ---------------------------------------------------------------------------
// Kernel C: per pixel, compute 2-4 edge weights, row-normalize, scatter into
// the (already zeroed) dense NxN matrix. Fully unrolled 4-direction loop with
// validity masks -> no indexed local arrays, no scratch.
// ---------------------------------------------------------------------------
__global__ void __launch_bounds__(256)
rwpl_scatter_kernel(const float* __restrict__ ws_flat,
                    const float* __restrict__ ws_scal,
                    float* __restrict__ P)
{
    const int p = blockIdx.x * blockDim.x + threadIdx.x;
    if (p >= NPIX) return;

    const float dmax     = ws_scal[0];
    const float dmin     = ws_scal[1];
    const float inv_dmax = 1.0f / (dmax + 1e-11f);
    const float e_sp     = __expf(-COEF);                       // spatial term: dist==1 always
    const float totalmax = e_sp + __expf(-COEF * dmin * inv_dmax);
    const float inv_tmax = 1.0f / (totalmax + 1e-11f);

    const int i = p / WW, j = p % WW;
    const float fp = ws_flat[p];

    const int  off[4]   = {-WW, WW, -1, 1};
    const bool valid[4] = {i > 0, i < HH - 1, j > 0, j < WW - 1};

    float w[4];
    float rowsum = 0.0f;
#pragma unroll
    for (int k = 0; k < 4; ++k) {
        const float fn  = valid[k] ? ws_flat[p + off[k]] : fp;
        const float d   = fabsf(fp - fn);
        const float tot = e_sp + __expf(-COEF * d * inv_dmax);
        const float wt  = valid[k] ? tot * inv_tmax : 0.0f;
        w[k] = wt;
        rowsum += wt;
    }
    const float inv_sum = 1.0f / rowsum;   // rowsum > 0 always (deg >= 2)
    const size_t base = (size_t)p * (size_t)NPIX;
#pragma unroll
    for (int k = 0; k < 4; ++k)
        if (valid[k])
            P[base + (size_t)(p + off[k])] = w[k] * inv_sum;
}

// ---------------------------------------------------------------------------
extern "C" void kernel_launch(void* const* d_in, const int* in_sizes, int n_in,
                              void* d_out, int out_size, void* d_ws, size_t ws_size,
                              hipStream_t stream)
{
    (void)in_sizes; (void)n_in; (void)ws_size;

    const float* img  = (const float*)d_in[0];   // image (96*96)
    const float* thre = (const float*)d_in[1];   // flat_boundary_thre (N)
    const float* addv = (const float*)d_in[2];   // flat_boundary_addition (N)
    float* out = (float*)d_out;                  // N*N floats

    float* ws_flat = (float*)d_ws;               // NPIX floats
    float* ws_scal = ws_flat + NPIX;             // 2 floats

    // A: flat_ + diff reductions (single WGP, async LDS pipeline)
    rwpl_flat_kernel<<<1, NT, 0, stream>>>(img, thre, addv, ws_flat, ws_scal);

    // B: zero-fill the dense output with NT 128-bit streaming stores
    const long long n4 = (long long)out_size / 4;   // 84,934,656 / 4
    rwpl_fill_kernel<<<8192, 256, 0, stream>>>((v4f*)out, n4);

    // C: scatter normalized edge weights
    rwpl_scatter_kernel<<<(NPIX + 255) / 256, 256, 0, stream>>>(ws_flat, ws_scal, out);
}